// DGM_d_2259152797867
// MI455X (gfx1250) — compile-verified
//
#include <hip/hip_runtime.h>
#include <hip/hip_bf16.h>

typedef __attribute__((ext_vector_type(2))) float v2f;
typedef __attribute__((ext_vector_type(8))) float v8f;

// Problem constants (from reference)
constexpr int BB    = 4;
constexpr int NN    = 4096;
constexpr int DD    = 256;   // D_IN == D_OUT
constexpr int KK    = 16;    // top-k
constexpr int LDSP  = 260;   // padded LDS row stride (floats) to spread banks

// -----------------------------------------------------------------------------
// Kernel 1: xe = x @ W  (fp32 WMMA 16x16x4), fused row-norm sq = sum(xe^2)
// grid = B*N/16 = 1024 workgroups of 256 threads (8 waves).
// Each workgroup: rows [wg*16, wg*16+16) of the flattened [B*N, 256] matrix,
// all 256 output columns (8 waves x 2 column tiles of 16).
// -----------------------------------------------------------------------------
__global__ __launch_bounds__(256) void xe_gemm_kernel(
    const float* __restrict__ x,   // [B*N, 256]
    const float* __restrict__ W,   // [256, 256]
    float* __restrict__ xe,        // [B*N, 256]  (d_out segment 0)
    float* __restrict__ sq)        // [B*N]       (workspace)
{
    __shared__ float xbuf[16 * LDSP];

    const int tid  = threadIdx.x;
    const int wave = tid >> 5;
    const int lane = tid & 31;
    const int half = lane >> 4;   // K sub-pair select for fp32 WMMA A/B frags
    const int l16  = lane & 15;
    const long rowbase = (long)blockIdx.x * 16;

    // Stage the 16 x-rows (16x256 f32) into LDS
    for (int i = tid; i < 16 * DD; i += 256) {
        const int r = i >> 8, c = i & 255;
        xbuf[r * LDSP + c] = x[(rowbase + r) * DD + c];
    }
    __syncthreads();

    v8f acc0 = {}, acc1 = {};
    for (int kk = 0; kk < DD; kk += 4) {
        const int ka = kk + half * 2;
        // A frag 16x4: lanes 0-15 hold (M=l16, K=ka,ka+1); lanes 16-31 K=ka+2,ka+3
        const v2f a = *(const v2f*)(xbuf + l16 * LDSP + ka);
        // B frag 4x16: lanes 0-15 hold (K=ka, N=col); lanes 16-31 K=ka+2
        const int col0 = wave * 32 + l16;
        v2f b0, b1;
        b0.x = W[(long)ka * DD + col0];
        b0.y = W[(long)(ka + 1) * DD + col0];
        b1.x = W[(long)ka * DD + col0 + 16];
        b1.y = W[(long)(ka + 1) * DD + col0 + 16];
        acc0 = __builtin_amdgcn_wmma_f32_16x16x4_f32(false, a, false, b0,
                                                     (short)0, acc0, false, false);
        acc1 = __builtin_amdgcn_wmma_f32_16x16x4_f32(false, a, false, b1,
                                                     (short)0, acc1, false, false);
    }
    __syncthreads();   // everyone done reading xbuf

    // Write xe to global and stash tile into LDS for deterministic row norms
#pragma unroll
    for (int g = 0; g < 8; ++g) {
        const int r = g + half * 8;
        const int c0 = wave * 32 + l16;
        const float v0 = acc0[g];
        const float v1 = acc1[g];
        xe[(rowbase + r) * DD + c0]      = v0;
        xe[(rowbase + r) * DD + c0 + 16] = v1;
        xbuf[r * LDSP + c0]      = v0;
        xbuf[r * LDSP + c0 + 16] = v1;
    }
    __syncthreads();

    if (tid < 16) {
        float s = 0.f;
        for (int c = 0; c < DD; ++c) {
            const float v = xbuf[tid * LDSP + c];
            s += v * v;
        }
        sq[rowbase + tid] = s;
    }
}

// -----------------------------------------------------------------------------
// Kernel 2: fused Gram (fp32 WMMA) + distance + Gumbel + streaming top-16.
// grid = B * N/16 = 1024 workgroups of 256 threads.
// Never materializes the [B,N,N] distance matrix: q is streamed once.
// -----------------------------------------------------------------------------
__global__ __launch_bounds__(256) void graph_topk_kernel(
    const float* __restrict__ xe,     // [B*N, 256]
    const float* __restrict__ sq,     // [B*N]
    const float* __restrict__ temp,   // [1]
    const float* __restrict__ q,      // [B, N, N]
    float* __restrict__ edges,        // [2, B*N*K]  (float-encoded ints)
    float* __restrict__ logprobs)     // [B*N*K]
{
    __shared__ float xbuf[16 * LDSP];   // row block of xe
    __shared__ float sqr[16];           // row norms
    __shared__ float cand[16 * 132];    // 128 candidates per row, padded
    __shared__ float topv[16][16];      // sorted ascending per row
    __shared__ int   topi[16][16];

    const int tid  = threadIdx.x;
    const int wave = tid >> 5;
    const int lane = tid & 31;
    const int half = lane >> 4;
    const int l16  = lane & 15;
    const int wg   = blockIdx.x;
    const int b    = wg >> 8;          // N/16 = 256 row-blocks per batch
    const int rb   = wg & 255;
    const long rowflat = (long)b * NN + (long)rb * 16;

    for (int i = tid; i < 16 * DD; i += 256) {
        const int r = i >> 8, c = i & 255;
        xbuf[r * LDSP + c] = xe[(rowflat + r) * DD + c];
    }
    if (tid < 16) {
        sqr[tid] = sq[rowflat + tid];
#pragma unroll
        for (int k = 0; k < KK; ++k) { topv[tid][k] = 3.0e38f; topi[tid][k] = 0; }
    }
    float tv = temp[0];
    tv = fminf(fmaxf(tv, -5.f), 5.f);
    const float tscale = expf(tv);

    const float* __restrict__ qb = q + (long)b * NN * NN;
    __syncthreads();

    for (int chunk = 0; chunk < NN / 128; ++chunk) {
        // Prefetch next chunk's q lines (global_prefetch_b8)
        if (chunk + 1 < NN / 128) {
            const int pr = tid & 15;          // row within block
            const int pc = (tid >> 4) * 8;    // spread over the 128 columns
            __builtin_prefetch(
                &qb[((long)(rb * 16 + pr)) * NN + (chunk + 1) * 128 + pc], 0, 0);
        }

        const int colbase = chunk * 128 + wave * 16;
        // This lane's B-column row in xe (column index colbase+l16)
        const float* __restrict__ xc = xe + ((long)b * NN + colbase + l16) * DD;

        v8f acc0 = {}, acc1 = {};
        for (int kk = 0; kk < DD; kk += 8) {
            const int ka = kk + half * 2;
            const v2f a0 = *(const v2f*)(xbuf + l16 * LDSP + ka);
            const v2f b0 = *(const v2f*)(xc + ka);
            acc0 = __builtin_amdgcn_wmma_f32_16x16x4_f32(false, a0, false, b0,
                                                         (short)0, acc0, false, false);
            const v2f a1 = *(const v2f*)(xbuf + l16 * LDSP + ka + 4);
            const v2f b1 = *(const v2f*)(xc + ka + 4);
            acc1 = __builtin_amdgcn_wmma_f32_16x16x4_f32(false, a1, false, b1,
                                                         (short)0, acc1, false, false);
        }

        // Distance + temperature + Gumbel perturbation, dump candidates to LDS
        const int   n   = colbase + l16;
        const float sqc = sq[(long)b * NN + n];
#pragma unroll
        for (int g = 0; g < 8; ++g) {
            const int r = g + half * 8;
            float d = sqr[r] + sqc - 2.0f * (acc0[g] + acc1[g]);
            d = fmaxf(d, 0.0f);
            const float qv = qb[((long)(rb * 16 + r)) * NN + n];
            const float lq = d * tscale - logf(-logf(qv));
            cand[r * 132 + wave * 16 + l16] = lq;
        }
        __syncthreads();

        // Per-row streaming top-16 (threshold filter -> rare sorted insertion)
        if (tid < 16) {
            const int r = tid;
            float thr = topv[r][KK - 1];
            for (int j = 0; j < 128; ++j) {
                const float v = cand[r * 132 + j];
                if (v < thr) {
                    int pos = KK - 1;
                    while (pos > 0 && topv[r][pos - 1] > v) {
                        topv[r][pos] = topv[r][pos - 1];
                        topi[r][pos] = topi[r][pos - 1];
                        --pos;
                    }
                    topv[r][pos] = v;
                    topi[r][pos] = chunk * 128 + j;
                    thr = topv[r][KK - 1];
                }
            }
        }
        __syncthreads();
    }

    // Emit logprobs (= -sorted lq) and edge rows, reference layout [2, B*N*K]
    if (tid < 16) {
        const int r = tid;
        const long nk = (rowflat + r) * KK;
#pragma unroll
        for (int k = 0; k < KK; ++k) {
            logprobs[nk + k] = -topv[r][k];
            edges[nk + k] = (float)(topi[r][k] + b * NN);                  // src idx
            edges[(long)BB * NN * KK + nk + k] = (float)(rb * 16 + r + b * NN); // row
        }
    }
}

// -----------------------------------------------------------------------------
// Launch
// -----------------------------------------------------------------------------
extern "C" void kernel_launch(void* const* d_in, const int* in_sizes, int n_in,
                              void* d_out, int out_size, void* d_ws, size_t ws_size,
                              hipStream_t stream) {
    const float* x    = (const float*)d_in[0];
    // d_in[1] = A (unused by reference)
    const float* W    = (const float*)d_in[2];
    const float* temp = (const float*)d_in[3];
    const float* q    = (const float*)d_in[4];

    float* out      = (float*)d_out;
    float* xe       = out;                                   // B*N*D  = 4194304
    float* edges    = out + (long)BB * NN * DD;              // 2*B*N*K = 524288
    float* logprobs = edges + (long)2 * BB * NN * KK;        // B*N*K  = 262144

    float* sq = (float*)d_ws;                                // B*N floats

    xe_gemm_kernel<<<BB * NN / 16, 256, 0, stream>>>(x, W, xe, sq);
    graph_topk_kernel<<<BB * NN / 16, 256, 0, stream>>>(xe, sq, temp, q,
                                                        edges, logprobs);
}